// FastenNet_49813030699353
// MI455X (gfx1250) — compile-verified
//
#include <hip/hip_runtime.h>
#include <hip/hip_bf16.h>
#include <math.h>

typedef __attribute__((ext_vector_type(2))) float v2f;
typedef __attribute__((ext_vector_type(8))) float v8f;

// ---------------------------------------------------------------------------
// Kernel 1: zero the layer-1 accumulator (ws is poisoned, we must clear it
// every call since we accumulate with atomics).
// ---------------------------------------------------------------------------
__global__ void rgcn_zero(float* __restrict__ p, int total) {
    int i = blockIdx.x * blockDim.x + threadIdx.x;
    if (i < total) p[i] = 0.0f;
}

// ---------------------------------------------------------------------------
// Kernel 2: layer-1 edge scatter.  msg1 = W1[r, src]; atomic segment-sum into
// acc[dst].  One thread per (edge, h) pair: 16 consecutive lanes read 16
// consecutive floats of W1 (coalesced 64B) and hit 16 consecutive atomic
// addresses of one h1 row.  Edge metadata loads are wave-broadcast.
// ---------------------------------------------------------------------------
__global__ void rgcn_layer1(const int* __restrict__ src,
                            const int* __restrict__ dst,
                            const int* __restrict__ etype,
                            const float* __restrict__ W1,
                            float* __restrict__ acc,
                            int E, int N) {
    long long idx = (long long)blockIdx.x * blockDim.x + threadIdx.x;
    long long total = (long long)E * 16;
    if (idx >= total) return;
    int e = (int)(idx >> 4);
    int h = (int)(idx & 15);
    int r = etype[e];
    int s = src[e];
    int d = dst[e];
    float v = W1[((size_t)r * (size_t)N + (size_t)s) * 16 + h];
    unsafeAtomicAdd(&acc[(size_t)d * 16 + h], v);
}

// ---------------------------------------------------------------------------
// Kernel 3: per-node finalize.  h1 = relu(acc + root1 + bias1) (back into ws),
// and initialize h2 = h1 @ root2 + bias2 directly in d_out.
// ---------------------------------------------------------------------------
__global__ void rgcn_node(const float* __restrict__ acc,
                          const float* __restrict__ root1,
                          const float* __restrict__ bias1,
                          const float* __restrict__ root2,
                          const float* __restrict__ bias2,
                          float* __restrict__ h1,
                          float* __restrict__ h2,
                          int N) {
    int i = blockIdx.x * blockDim.x + threadIdx.x;
    if (i >= N) return;
    float h[16];
#pragma unroll
    for (int k = 0; k < 16; ++k) {
        float v = acc[(size_t)i * 16 + k] + root1[(size_t)i * 16 + k] + bias1[k];
        h[k] = v > 0.0f ? v : 0.0f;
    }
#pragma unroll
    for (int k = 0; k < 16; ++k) h1[(size_t)i * 16 + k] = h[k];
#pragma unroll
    for (int c = 0; c < 16; ++c) {
        float s = bias2[c];
#pragma unroll
        for (int k = 0; k < 16; ++k) s += h[k] * root2[k * 16 + c];
        h2[(size_t)i * 16 + c] = s;
    }
}

// ---------------------------------------------------------------------------
// Kernel 4: layer-2 edge messages via WMMA.  One wave handles 16 edges.
// Edges are sorted by relation, so the chunk is (almost always) one relation:
//   A (16x16, f32) = h1 rows gathered at src[e0..e0+15]
//   B (16x16, f32) = W2[r]
//   D = A @ B  computed as 4 chained V_WMMA_F32_16X16X4_F32 (K=4 slices)
// then D rows are atomically scattered to h2[dst].
// A layout (32-bit, 16x4): lanes 0-15 = M rows, lane-half selects K-pair,
// VGPR selects K within the pair; B mirrors this with N across lanes.
// Rare mixed-relation / tail chunks fall back to a scalar matvec path
// (wave-uniform branch, so EXEC is all-ones for the WMMA path as required).
// ---------------------------------------------------------------------------
__global__ __launch_bounds__(128) void rgcn_layer2(const int* __restrict__ src,
                                                   const int* __restrict__ dst,
                                                   const int* __restrict__ etype,
                                                   const float* __restrict__ h1,
                                                   const float* __restrict__ W2,
                                                   float* __restrict__ h2,
                                                   int E, int nchunks) {
    int wave = (blockIdx.x * blockDim.x + threadIdx.x) >> 5;
    int lane = threadIdx.x & 31;
    if (wave >= nchunks) return;
    int e0 = wave * 16;
    int cnt = E - e0;
    if (cnt > 16) cnt = 16;

    bool full = (cnt == 16);
    int rFirst = etype[e0];
    bool uniform = full && (etype[e0 + 15] == rFirst);   // wave-uniform condition

    if (uniform) {
        const float* W2r = W2 + (size_t)rFirst * 256;    // 16x16 row-major [h][c]
        int m  = lane & 15;       // A: edge-in-chunk row   B/D: class column
        int hi = lane >> 4;       // lane-half -> K-pair select
        int s  = src[e0 + m];
        const float* arow = h1 + (size_t)s * 16 + hi * 2;

        v8f c = {0.f, 0.f, 0.f, 0.f, 0.f, 0.f, 0.f, 0.f};
#pragma unroll
        for (int k = 0; k < 4; ++k) {
            v2f a = *(const v2f*)(arow + 4 * k);          // A[m][4k+2hi .. +1]
            int kb = 4 * k + hi * 2;
            v2f b;
            b.x = W2r[(kb + 0) * 16 + m];                 // B[kb  ][m]
            b.y = W2r[(kb + 1) * 16 + m];                 // B[kb+1][m]
            c = __builtin_amdgcn_wmma_f32_16x16x4_f32(
                    /*neg_a=*/false, a, /*neg_b=*/false, b,
                    /*c_mod=*/(short)0, c, /*reuse_a=*/false, /*reuse_b=*/false);
        }
        // D: VGPR j holds row (j + 8*hi), column m. Scatter with f32 atomics.
#pragma unroll
        for (int j = 0; j < 8; ++j) {
            int medge = j + hi * 8;
            int d = dst[e0 + medge];
            unsafeAtomicAdd(&h2[(size_t)d * 16 + m], c[j]);
        }
    } else {
        // mixed-relation boundary chunk or tail: scalar matvec per edge
        if (lane < cnt) {
            int e = e0 + lane;
            int r = etype[e];
            int s = src[e];
            int d = dst[e];
            const float* hrow = h1 + (size_t)s * 16;
            const float* W2r  = W2 + (size_t)r * 256;
            float hreg[16];
#pragma unroll
            for (int k = 0; k < 16; ++k) hreg[k] = hrow[k];
#pragma unroll
            for (int cc = 0; cc < 16; ++cc) {
                float accv = 0.0f;
#pragma unroll
                for (int k = 0; k < 16; ++k) accv += hreg[k] * W2r[k * 16 + cc];
                unsafeAtomicAdd(&h2[(size_t)d * 16 + cc], accv);
            }
        }
    }
}

// ---------------------------------------------------------------------------
// Kernel 5: in-place row log_softmax over C=16.
// ---------------------------------------------------------------------------
__global__ void rgcn_logsoftmax(float* __restrict__ out, int N) {
    int i = blockIdx.x * blockDim.x + threadIdx.x;
    if (i >= N) return;
    float* row = out + (size_t)i * 16;
    float v[16];
    float m = -INFINITY;
#pragma unroll
    for (int c = 0; c < 16; ++c) { v[c] = row[c]; m = fmaxf(m, v[c]); }
    float s = 0.0f;
#pragma unroll
    for (int c = 0; c < 16; ++c) s += expf(v[c] - m);
    float ls = logf(s);
#pragma unroll
    for (int c = 0; c < 16; ++c) row[c] = v[c] - m - ls;
}

// ---------------------------------------------------------------------------
extern "C" void kernel_launch(void* const* d_in, const int* in_sizes, int n_in,
                              void* d_out, int out_size, void* d_ws, size_t ws_size,
                              hipStream_t stream) {
    const int*   edge_index = (const int*)d_in[0];   // (2, E)
    const int*   edge_type  = (const int*)d_in[1];   // (E,)
    // d_in[2] tensor_slice unused (chunking handles relation boundaries)
    const float* W1    = (const float*)d_in[3];      // (R, N, 16)
    const float* root1 = (const float*)d_in[4];      // (N, 16)
    const float* bias1 = (const float*)d_in[5];      // (16,)
    const float* W2    = (const float*)d_in[6];      // (R, 16, 16)
    const float* root2 = (const float*)d_in[7];      // (16, 16)
    const float* bias2 = (const float*)d_in[8];      // (16,)
    float* out = (float*)d_out;                      // (N, 16)

    const int E = in_sizes[1];
    const int N = in_sizes[4] / 16;

    const int* src = edge_index;        // row 0
    const int* dst = edge_index + E;    // row 1

    float* h1acc = (float*)d_ws;        // N*16 floats: acc, then relu'd h1

    // 1) zero accumulator
    {
        int total = N * 16;
        int blocks = (total + 255) / 256;
        rgcn_zero<<<blocks, 256, 0, stream>>>(h1acc, total);
    }
    // 2) layer-1 edge scatter
    {
        long long total = (long long)E * 16;
        int blocks = (int)((total + 255) / 256);
        rgcn_layer1<<<blocks, 256, 0, stream>>>(src, dst, edge_type, W1, h1acc, E, N);
    }
    // 3) node finalize: h1 (ws, in place) + h2 init (d_out)
    {
        int blocks = (N + 255) / 256;
        rgcn_node<<<blocks, 256, 0, stream>>>(h1acc, root1, bias1, root2, bias2,
                                              h1acc, out, N);
    }
    // 4) layer-2 WMMA edge messages: 1 wave per 16-edge chunk, 4 waves/block
    {
        int nchunks = (E + 15) / 16;
        int blocks = (nchunks + 3) / 4;
        rgcn_layer2<<<blocks, 128, 0, stream>>>(src, dst, edge_type, h1acc, W2,
                                                out, E, nchunks);
    }
    // 5) log_softmax in place
    {
        int blocks = (N + 255) / 256;
        rgcn_logsoftmax<<<blocks, 256, 0, stream>>>(out, N);
    }
}